// DIOU3DLoss_19215683682795
// MI455X (gfx1250) — compile-verified
//
#include <hip/hip_runtime.h>
#include <hip/hip_bf16.h>

typedef __attribute__((ext_vector_type(2))) float v2f;
typedef __attribute__((ext_vector_type(8))) float v8f;

#define TPB   128
#define EPSI  1e-8f

// ---------------------------------------------------------------------------
// Exact f32 block reduction using V_WMMA_F32_16X16X4_F32.
// A = all-ones (16x4) => every row of D equals the column sums of B, so the
// result is independent of the exact (k,n) lane layout of B. VGPR0 of D holds
// 32 cells covering each of the 16 columns exactly twice (rows M=0 and M=8),
// so 0.5 * sum(lane values of acc[0]) == sum of all B elements fed in.
// Returns the block total on tid==0 (undefined elsewhere).
// ---------------------------------------------------------------------------
__device__ __forceinline__ float block_sum_wmma(float val, float* s_red, int tid)
{
    s_red[tid] = val;
    __syncthreads();
    if (tid < 32) {                       // wave 0, EXEC all ones (wave32)
        v2f ones = {1.0f, 1.0f};
        v8f acc  = {};
#pragma unroll
        for (int c = 0; c < TPB / 64; ++c) {
            v2f b;
            b.x = s_red[c * 64 + tid];
            b.y = s_red[c * 64 + 32 + tid];
            acc = __builtin_amdgcn_wmma_f32_16x16x4_f32(
                      false, ones, false, b, (short)0, acc, false, false);
        }
        s_red[tid] = acc[0];
    }
    __syncthreads();
    float total = 0.0f;
    if (tid == 0) {
        float s = 0.0f;
#pragma unroll
        for (int i = 0; i < 32; ++i) s += s_red[i];
        total = 0.5f * s;
    }
    return total;
}

// ---------------------------------------------------------------------------
// One thread per box pair. LDS-resident 24-vertex angular sort.
// ---------------------------------------------------------------------------
__global__ void __launch_bounds__(TPB)
diou3d_loss_kernel(const float* __restrict__ pred,
                   const float* __restrict__ targ,
                   float* __restrict__ partial, int n)
{
    __shared__ float s_ang[TPB * 25];
    __shared__ float s_vx [TPB * 25];
    __shared__ float s_vy [TPB * 25];
    __shared__ float s_red[TPB];

    const int tid = threadIdx.x;
    const int idx = blockIdx.x * TPB + tid;

    float* ang = &s_ang[tid * 25];
    float* wx  = &s_vx [tid * 25];
    float* wy  = &s_vy [tid * 25];

    float loss = 0.0f;
    if (idx < n) {
        float p[7], t[7];
#pragma unroll
        for (int i = 0; i < 7; ++i) { p[i] = pred[idx * 7 + i]; t[i] = targ[idx * 7 + i]; }

        // ---------------- 2D rotated corners ----------------
        const float X4[4] = { 0.5f, -0.5f, -0.5f,  0.5f };
        const float Y4[4] = { 0.5f,  0.5f, -0.5f, -0.5f };
        float c1a = cosf(p[6]), s1a = sinf(p[6]);
        float c2a = cosf(t[6]), s2a = sinf(t[6]);
        float c1x[4], c1y[4], c2x[4], c2y[4];
#pragma unroll
        for (int k = 0; k < 4; ++k) {
            float xx = X4[k] * p[3], yy = Y4[k] * p[4];
            c1x[k] = xx * c1a - yy * s1a + p[0];
            c1y[k] = xx * s1a + yy * c1a + p[1];
            xx = X4[k] * t[3]; yy = Y4[k] * t[4];
            c2x[k] = xx * c2a - yy * s2a + t[0];
            c2y[k] = xx * s2a + yy * c2a + t[1];
        }

        float sumx = 0.0f, sumy = 0.0f, nvf = 0.0f;

        // ---------------- corners-in-other-box (vertices 0..7) ----------------
        const float TOL = 1e-6f;
        float abx = c2x[1] - c2x[0], aby = c2y[1] - c2y[0];
        float adx = c2x[3] - c2x[0], ady = c2y[3] - c2y[0];
        float ab2 = abx * abx + aby * aby, ad2 = adx * adx + ady * ady;
        float ebx = c1x[1] - c1x[0], eby = c1y[1] - c1y[0];
        float edx = c1x[3] - c1x[0], edy = c1y[3] - c1y[0];
        float eb2 = ebx * ebx + eby * eby, ed2 = edx * edx + edy * edy;
#pragma unroll
        for (int k = 0; k < 4; ++k) {
            float amx = c1x[k] - c2x[0], amy = c1y[k] - c2y[0];
            float pab = (abx * amx + aby * amy) / ab2;
            float pad = (adx * amx + ady * amy) / ad2;
            bool m = (pab > -TOL) && (pab < 1.0f + TOL) && (pad > -TOL) && (pad < 1.0f + TOL);
            wx[k] = c1x[k]; wy[k] = c1y[k]; ang[k] = m ? 1.0f : 0.0f;
            if (m) { sumx += c1x[k]; sumy += c1y[k]; nvf += 1.0f; }

            amx = c2x[k] - c1x[0]; amy = c2y[k] - c1y[0];
            pab = (ebx * amx + eby * amy) / eb2;
            pad = (edx * amx + edy * amy) / ed2;
            m = (pab > -TOL) && (pab < 1.0f + TOL) && (pad > -TOL) && (pad < 1.0f + TOL);
            wx[4 + k] = c2x[k]; wy[4 + k] = c2y[k]; ang[4 + k] = m ? 1.0f : 0.0f;
            if (m) { sumx += c2x[k]; sumy += c2y[k]; nvf += 1.0f; }
        }

        // ---------------- edge-edge intersections (vertices 8..23) -------------
#pragma unroll
        for (int i = 0; i < 4; ++i) {
            float x1 = c1x[i], y1 = c1y[i];
            float x2 = c1x[(i + 1) & 3], y2 = c1y[(i + 1) & 3];
#pragma unroll
            for (int j = 0; j < 4; ++j) {
                float x3 = c2x[j], y3 = c2y[j];
                float x4 = c2x[(j + 1) & 3], y4 = c2y[(j + 1) & 3];
                float num  = (y4 - y3) * (x2 - x1) - (x4 - x3) * (y2 - y1);
                float dent = (x4 - x3) * (y1 - y3) - (y4 - y3) * (x1 - x3);
                float denu = (x2 - x1) * (y1 - y3) - (y2 - y1) * (x1 - x3);
                bool m = false;
                if (num != 0.0f) {
                    float tt = dent / num;
                    float uu = -denu / num;
                    m = (tt > 0.0f) && (tt < 1.0f) && (uu > 0.0f) && (uu < 1.0f);
                }
                float t2 = dent / (num + EPSI);
                float ix = m ? (x1 + t2 * (x2 - x1)) : 0.0f;
                float iy = m ? (y1 + t2 * (y2 - y1)) : 0.0f;
                int k = 8 + i * 4 + j;
                wx[k] = ix; wy[k] = iy; ang[k] = m ? 1.0f : 0.0f;
                if (m) { sumx += ix; sumy += iy; nvf += 1.0f; }
            }
        }

        // ---------------- polygon area (sort-by-angle + shoelace) --------------
        float nv = fmaxf(nvf, 1.0f);
        float mx = sumx / nv, my = sumy / nv;
#pragma unroll
        for (int k = 0; k < 24; ++k) {
            float vx_ = wx[k] - mx, vy_ = wy[k] - my;
            wx[k] = vx_; wy[k] = vy_;
            ang[k] = (ang[k] != 0.0f) ? atan2f(vy_, vx_) : (1.0e6f + (float)k);
        }
        // stable insertion sort (masked entries carry distinct keys 1e6+k)
        for (int i = 1; i < 24; ++i) {
            float a = ang[i], x = wx[i], y = wy[i];
            int j = i - 1;
            while (j >= 0 && ang[j] > a) {
                ang[j + 1] = ang[j]; wx[j + 1] = wx[j]; wy[j + 1] = wy[j];
                --j;
            }
            ang[j + 1] = a; wx[j + 1] = x; wy[j + 1] = y;
        }
        float fx = wx[0], fy = wy[0];          // first = sorted[0] (pre-replacement)
        float px = fx, py = fy, sh = 0.0f;
#pragma unroll
        for (int k = 1; k < 24; ++k) {
            float qx, qy;
            if (ang[k] < 1.0e5f) { qx = wx[k]; qy = wy[k]; }
            else                 { qx = fx;    qy = fy;    }
            sh += px * qy - py * qx;
            px = qx; py = qy;
        }
        sh += px * fy - py * fx;               // wrap to element 0
        float inter2d = 0.5f * fabsf(sh);

        // ---------------- z overlap, volumes, IoU ----------------
        float zmax1 = p[2] + 0.5f * p[5], zmin1 = p[2] - 0.5f * p[5];
        float zmax2 = t[2] + 0.5f * t[5], zmin2 = t[2] - 0.5f * t[5];
        float zov = fmaxf(fminf(zmax1, zmax2) - fmaxf(zmin1, zmin2), 0.0f);
        float inter3 = inter2d * zov;
        float vol1 = p[3] * p[4] * p[5], vol2 = t[3] * t[4] * t[5];
        float iou = inter3 / (vol1 + vol2 - inter3);
        if (!(iou == iou)) iou = 0.0f;         // nan_to_num(nan=0)

        // ---------------- center + 3D-corner distances ----------------
        float dx0 = p[0] - t[0], dy0 = p[1] - t[1], dz0 = p[2] - t[2];
        float ctd = dx0 * dx0 + dy0 * dy0 + dz0 * dz0;

        const float CNX[8] = {-0.5f,-0.5f,-0.5f,-0.5f, 0.5f, 0.5f, 0.5f, 0.5f};
        const float CNY[8] = {-0.5f,-0.5f, 0.5f, 0.5f,-0.5f,-0.5f, 0.5f, 0.5f};
        const float CNZ[8] = {-0.5f, 0.5f, 0.5f,-0.5f,-0.5f, 0.5f, 0.5f,-0.5f};
        float cacc = 0.0f;
#pragma unroll
        for (int k = 0; k < 8; ++k) {
            float ax = CNX[k] * p[3], ay = CNY[k] * p[4], az = CNZ[k] * p[5];
            float gx =  ax * c1a + az * s1a + p[0];
            float gy =  ay + p[1];
            float gz = -ax * s1a + az * c1a + p[2];
            float bx = CNX[k] * t[3], by = CNY[k] * t[4], bz = CNZ[k] * t[5];
            float hx =  bx * c2a + bz * s2a + t[0];
            float hy =  by + t[1];
            float hz = -bx * s2a + bz * c2a + t[2];
            float ddx = gx - hx, ddy = gy - hy, ddz = gz - hz;
            cacc += ddx * ddx + ddy * ddy + ddz * ddz;
        }
        float cnd  = cacc * 0.125f;
        float did  = t[3] * t[3] + t[4] * t[4] + t[5] * t[5];
        float num2 = ctd + cnd;
        loss = 1.0f - iou + num2 / (num2 + did + 1.0e-6f);
    }

    float bs = block_sum_wmma(loss, s_red, tid);
    if (tid == 0) partial[blockIdx.x] = bs;
}

// ---------------------------------------------------------------------------
// Single-block finish: sum partials, scale by 1/N.
// ---------------------------------------------------------------------------
__global__ void __launch_bounds__(TPB)
final_reduce_kernel(const float* __restrict__ partial, float* __restrict__ out,
                    int nparts, float invn)
{
    __shared__ float s_red[TPB];
    const int tid = threadIdx.x;
    float s = 0.0f;
    for (int i = tid; i < nparts; i += TPB) s += partial[i];
    float tot = block_sum_wmma(s, s_red, tid);
    if (tid == 0) out[0] = tot * invn;   // LOSS_WEIGHT == 1.0
}

extern "C" void kernel_launch(void* const* d_in, const int* in_sizes, int n_in,
                              void* d_out, int out_size, void* d_ws, size_t ws_size,
                              hipStream_t stream)
{
    (void)n_in; (void)out_size; (void)ws_size;
    const float* pred = (const float*)d_in[0];
    const float* targ = (const float*)d_in[1];
    int n = in_sizes[0] / 7;                       // (N, 7) boxes
    int blocks = (n + TPB - 1) / TPB;              // 2048 for N = 262144
    float* partial = (float*)d_ws;                 // blocks * 4 bytes of scratch

    diou3d_loss_kernel<<<blocks, TPB, 0, stream>>>(pred, targ, partial, n);
    final_reduce_kernel<<<1, TPB, 0, stream>>>(partial, (float*)d_out, blocks,
                                               1.0f / (float)n);
}